// UtoGlobal_55087250538899
// MI455X (gfx1250) — compile-verified
//
#include <hip/hip_runtime.h>

// Problem constants (from reference): N=100000, C=32, F=128, OUT=128, B=512
#define C_   32
#define F_   128
#define OUT_ 128
#define B_   512

#define NODES_PER_WAVE  16
#define WAVES_PER_BLOCK 8

typedef __attribute__((ext_vector_type(2))) float v2f;
typedef __attribute__((ext_vector_type(4))) float v4f;
typedef __attribute__((ext_vector_type(8))) float v8f;

// ---------------------------------------------------------------------------
// Pass 1: stream u (1.64 GB), produce per-graph sums of:
//   diag slice u[n, coloring[n], :]        -> sdiag[B][F]
//   per-node mean (sum_c u[n,c,:])/nb[n]   -> smean[B][F]
// and per-graph node counts -> cnt[B].
// One wave handles NODES_PER_WAVE consecutive nodes; lane owns 4 features.
// batch is sorted, so we accumulate in registers and flush on graph change.
// ---------------------------------------------------------------------------
__global__ __launch_bounds__(256)
void utog_pass1(const float* __restrict__ u,
                const int*   __restrict__ coloring,
                const int*   __restrict__ batch,
                const float* __restrict__ nbatch,
                float* __restrict__ sdiag,
                float* __restrict__ smean,
                float* __restrict__ cnt,
                int n_nodes)
{
    const int wave = blockIdx.x * WAVES_PER_BLOCK + (threadIdx.x >> 5);
    const int lane = threadIdx.x & 31;
    int n0 = wave * NODES_PER_WAVE;
    if (n0 >= n_nodes) return;
    int n1 = n0 + NODES_PER_WAVE;
    if (n1 > n_nodes) n1 = n_nodes;

    const int f0 = lane * 4;                 // this lane's 4 features
    float accD[4] = {0.f, 0.f, 0.f, 0.f};
    float accM[4] = {0.f, 0.f, 0.f, 0.f};
    int curB = -1;

    for (int n = n0; n < n1; ++n) {
        const int b = batch[n];
        if (b != curB) {
            if (curB >= 0) {
                #pragma unroll
                for (int j = 0; j < 4; ++j) {
                    atomicAdd(&sdiag[curB * F_ + f0 + j], accD[j]);
                    atomicAdd(&smean[curB * F_ + f0 + j], accM[j]);
                    accD[j] = 0.f; accM[j] = 0.f;
                }
            }
            curB = b;
        }
        const float nb = nbatch[n];          // counts[batch[n]] (>=1)
        if (lane == 0) cnt[b] = nb;          // same value from all writers
        const float rinv = 1.0f / nb;

        const float* base = u + (size_t)n * (C_ * F_) + f0;
        float s0 = 0.f, s1 = 0.f, s2 = 0.f, s3 = 0.f;
        #pragma unroll
        for (int c = 0; c < C_; ++c) {       // 32 independent b128 loads
            v4f v = *(const v4f*)(base + c * F_);
            s0 += v.x; s1 += v.y; s2 += v.z; s3 += v.w;
        }
        const int col = coloring[n];
        v4f d = *(const v4f*)(base + col * F_);   // row is cache-hot

        accM[0] += s0 * rinv; accM[1] += s1 * rinv;
        accM[2] += s2 * rinv; accM[3] += s3 * rinv;
        accD[0] += d.x; accD[1] += d.y; accD[2] += d.z; accD[3] += d.w;
    }
    if (curB >= 0) {
        #pragma unroll
        for (int j = 0; j < 4; ++j) {
            atomicAdd(&sdiag[curB * F_ + f0 + j], accD[j]);
            atomicAdd(&smean[curB * F_ + f0 + j], accM[j]);
        }
    }
}

// ---------------------------------------------------------------------------
// Pass 2: finalize means and do the dual GEMM with V_WMMA_F32_16X16X4_F32.
//   T = sdiag/cnt, S = smean/cnt - T
//   out = T @ W1^T + S @ W2^T + b1 + b2          (512 x 128)
// 256 tiles of 16x16; one full wave per tile (EXEC all ones, no divergence).
// A frag (16x4):  lanes 0-15 -> K={0,1}, lanes 16-31 -> K={2,3}, M = lane&15
// B frag (4x16):  vgpr0 -> K=0(+2*hi), vgpr1 -> K=1(+2*hi), N = lane&15
// C/D (16x16):    vgpr r -> M = r + 8*hi, N = lane&15
// ---------------------------------------------------------------------------
__global__ __launch_bounds__(256)
void utog_pass2(const float* __restrict__ sdiag,
                const float* __restrict__ smean,
                const float* __restrict__ cnt,
                const float* __restrict__ W1,
                const float* __restrict__ b1,
                const float* __restrict__ W2,
                const float* __restrict__ b2,
                float* __restrict__ out)
{
    const int wave = blockIdx.x * WAVES_PER_BLOCK + (threadIdx.x >> 5); // 0..255
    const int lane = threadIdx.x & 31;
    const int mt   = wave >> 3;          // 0..31  (batch tile)
    const int nt   = wave & 7;           // 0..7   (out-feature tile)
    const int row  = lane & 15;
    const int hi   = lane >> 4;          // 0 or 1

    const int bRow = mt * 16 + row;      // A-matrix row = batch index
    const int oCol = nt * 16 + row;      // B-matrix column = out feature

    const float inv  = 1.0f / fmaxf(cnt[bRow], 1.0f);
    const float bias = b1[oCol] + b2[oCol];

    const float* sdRow = sdiag + bRow * F_ + hi * 2;
    const float* smRow = smean + bRow * F_ + hi * 2;
    const float* w1Row = W1    + oCol * F_ + hi * 2;
    const float* w2Row = W2    + oCol * F_ + hi * 2;

    v8f acc = {};
    #pragma unroll 4
    for (int k = 0; k < F_; k += 4) {
        v2f sd = *(const v2f*)(sdRow + k);
        v2f sm = *(const v2f*)(smRow + k);
        v2f w1 = *(const v2f*)(w1Row + k);
        v2f w2 = *(const v2f*)(w2Row + k);
        v2f t;  t.x = sd.x * inv;        t.y = sd.y * inv;
        v2f s;  s.x = sm.x * inv - t.x;  s.y = sm.y * inv - t.y;
        acc = __builtin_amdgcn_wmma_f32_16x16x4_f32(
                  false, t, false, w1, (short)0, acc, false, false);
        acc = __builtin_amdgcn_wmma_f32_16x16x4_f32(
                  false, s, false, w2, (short)0, acc, false, false);
    }

    const int Mbase = mt * 16 + 8 * hi;
    #pragma unroll
    for (int r = 0; r < 8; ++r) {
        out[(Mbase + r) * OUT_ + oCol] = acc[r] + bias;
    }
}

// ---------------------------------------------------------------------------
extern "C" void kernel_launch(void* const* d_in, const int* in_sizes, int n_in,
                              void* d_out, int out_size, void* d_ws, size_t ws_size,
                              hipStream_t stream)
{
    const float* u        = (const float*)d_in[0];
    const int*   coloring = (const int*)  d_in[1];
    const int*   batch    = (const int*)  d_in[2];
    const float* nbatch   = (const float*)d_in[3];
    const float* W1       = (const float*)d_in[4];
    const float* b1       = (const float*)d_in[5];
    const float* W2       = (const float*)d_in[6];
    const float* b2       = (const float*)d_in[7];
    float*       out      = (float*)d_out;

    const int n_nodes = in_sizes[2];       // batch has N elements

    float* sdiag = (float*)d_ws;           // B*F
    float* smean = sdiag + B_ * F_;        // B*F
    float* cnt   = smean + B_ * F_;        // B

    const size_t zero_bytes = (size_t)(2 * B_ * F_ + B_) * sizeof(float);
    hipMemsetAsync(d_ws, 0, zero_bytes, stream);

    const int waves  = (n_nodes + NODES_PER_WAVE - 1) / NODES_PER_WAVE;
    const int blocks = (waves + WAVES_PER_BLOCK - 1) / WAVES_PER_BLOCK;
    utog_pass1<<<blocks, 256, 0, stream>>>(u, coloring, batch, nbatch,
                                           sdiag, smean, cnt, n_nodes);

    // 256 waves = 32 blocks x 8 waves; exactly one wave per 16x16 output tile
    utog_pass2<<<32, 256, 0, stream>>>(sdiag, smean, cnt, W1, b1, W2, b2, out);
}